// MultiDetector_87033217286358
// MI455X (gfx1250) — compile-verified
//
#include <hip/hip_runtime.h>

typedef float f4  __attribute__((ext_vector_type(4)));
typedef float f2  __attribute__((ext_vector_type(2)));
typedef float v8f __attribute__((ext_vector_type(8)));

// Problem constants (fixed by the reference)
#define BATCH 256
#define CHAN  2048
#define POOLN 64          // T*H*W = 16*2*2, contiguous innermost
#define NLOC  2
#define NCONF 3
#define NPAD  16          // N dimension padded to one WMMA tile

// ---------------------------------------------------------------------------
// Stage 0: pack [W_loc; W_conf; zeros] into Wpad[16][2048] so the GEMM loop
// needs no predication (keeps EXEC all-ones around the WMMAs, no exec
// save/restore or zero-fill movs in the hot loop).
// ---------------------------------------------------------------------------
__global__ __launch_bounds__(256) void md_packw_kernel(
    const float* __restrict__ W_loc, const float* __restrict__ W_conf,
    float* __restrict__ Wpad) {
  const int i   = blockIdx.x * 256 + threadIdx.x;  // 0 .. 16*2048-1
  const int row = i >> 11;                          // / CHAN
  const int col = i & (CHAN - 1);
  float v = 0.0f;
  if (row < NLOC)              v = W_loc[row * CHAN + col];
  else if (row < NLOC + NCONF) v = W_conf[(row - NLOC) * CHAN + col];
  Wpad[i] = v;
}

// ---------------------------------------------------------------------------
// Stage 1: pooled[b][c] = mean of 64 contiguous floats.
// One float4 NT load per thread, 16 lanes cooperate per (b,c) group.
// 8,388,608 float4s total -> 32768 blocks x 256 threads, fully coalesced.
// ---------------------------------------------------------------------------
__global__ __launch_bounds__(256) void md_pool_kernel(
    const float* __restrict__ x, float* __restrict__ pooled) {
  const unsigned t = blockIdx.x * 256u + threadIdx.x;      // float4 index
  const f4 v = __builtin_nontemporal_load((const f4*)x + t);
  float s = (v.x + v.y) + (v.z + v.w);
  // reduce within each 16-lane half of the wave32 (xor never crosses bit 4)
  s += __shfl_xor(s, 8, 32);
  s += __shfl_xor(s, 4, 32);
  s += __shfl_xor(s, 2, 32);
  s += __shfl_xor(s, 1, 32);
  if ((threadIdx.x & 15u) == 0u)
    pooled[t >> 4] = s * (1.0f / (float)POOLN);
}

// ---------------------------------------------------------------------------
// Stage 2: out = pooled @ Wpad^T + bias, via V_WMMA_F32_16X16X4_F32.
// One wave per 16-row tile (16 blocks x 32 threads). Full fp32 precision.
//
// Fragment layouts (CDNA5 ISA 7.12.2, 32-bit operands):
//   A 16x4 : lane<16 -> M=lane, {v0,v1}={K0,K1}; lane>=16 -> M=lane-16, {K2,K3}
//   B 4x16 : lane<16 -> N=lane, {v0,v1}={K0,K1}; lane>=16 -> N=lane-16, {K2,K3}
//   C/D 16x16 : VGPR r, lane<16 -> M=r, N=lane; lane>=16 -> M=r+8, N=lane-16
// ---------------------------------------------------------------------------
__global__ __launch_bounds__(32) void md_head_kernel(
    const float* __restrict__ pooled, const float* __restrict__ Wpad,
    const float* __restrict__ b_loc,  const float* __restrict__ b_conf,
    float* __restrict__ out) {
  const int lane = threadIdx.x;       // 0..31, all active (EXEC all-ones)
  const int nm   = lane & 15;         // row (A) / col (B) index of this lane
  const int koff = (lane >> 4) * 2;   // K sub-offset: {0,1} or {2,3}
  const int tile = blockIdx.x;        // 16 tiles of 16 rows each

  const float* arow = pooled + (size_t)(tile * 16 + nm) * CHAN + koff;
  const float* wrow = Wpad   + (size_t)nm * CHAN + koff;

  v8f acc0 = {};
  v8f acc1 = {};
#pragma unroll 4
  for (int k = 0; k < CHAN; k += 8) {
    const f2 a0 = *(const f2*)(arow + k);
    const f2 a1 = *(const f2*)(arow + k + 4);
    const f2 b0 = *(const f2*)(wrow + k);
    const f2 b1 = *(const f2*)(wrow + k + 4);
    // v_wmma_f32_16x16x4_f32: 8 args (neg_a, A, neg_b, B, c_mod, C, reuseA, reuseB)
    acc0 = __builtin_amdgcn_wmma_f32_16x16x4_f32(
        false, a0, false, b0, (short)0, acc0, false, false);
    acc1 = __builtin_amdgcn_wmma_f32_16x16x4_f32(
        false, a1, false, b1, (short)0, acc1, false, false);
  }
  const v8f d = acc0 + acc1;

  // Bias + scatter: loc -> out[0:512] (b*2+j), conf -> out[512:1280] (b*3+j)
  const int mbase = tile * 16 + (lane >> 4) * 8;
#pragma unroll
  for (int r = 0; r < 8; ++r) {
    const int   row = mbase + r;
    const float val = d[r];
    if (nm < NLOC)
      out[row * NLOC + nm] = val + b_loc[nm];
    else if (nm < NLOC + NCONF)
      out[BATCH * NLOC + row * NCONF + (nm - NLOC)] = val + b_conf[nm - NLOC];
  }
}

// ---------------------------------------------------------------------------
// Inputs: 0=x, 1=start_boundaries(unused), 2=W_loc, 3=b_loc, 4=W_conf, 5=b_conf
// Output: 1280 floats = loc[256,2] ++ conf[256,3]
// Workspace: pooled[256][2048] (2 MiB) then Wpad[16][2048] (128 KiB).
// ---------------------------------------------------------------------------
extern "C" void kernel_launch(void* const* d_in, const int* in_sizes, int n_in,
                              void* d_out, int out_size, void* d_ws, size_t ws_size,
                              hipStream_t stream) {
  const float* x      = (const float*)d_in[0];
  const float* W_loc  = (const float*)d_in[2];
  const float* b_loc  = (const float*)d_in[3];
  const float* W_conf = (const float*)d_in[4];
  const float* b_conf = (const float*)d_in[5];
  float* pooled = (float*)d_ws;
  float* Wpad   = pooled + (size_t)BATCH * CHAN;
  float* out    = (float*)d_out;

  md_packw_kernel<<<(NPAD * CHAN) / 256, 256, 0, stream>>>(W_loc, W_conf, Wpad);
  // 256*2048*64 floats / 4 per thread / 256 per block = 32768 blocks
  md_pool_kernel<<<(BATCH * CHAN * POOLN) / 4 / 256, 256, 0, stream>>>(x, pooled);
  md_head_kernel<<<BATCH / 16, 32, 0, stream>>>(pooled, Wpad, b_loc, b_conf, out);
}